// Sampler_1382979469918
// MI455X (gfx1250) — compile-verified
//
#include <hip/hip_runtime.h>
#include <stdint.h>
#include <math.h>

#define NT 512
#define TILE_ELEMS (NT * 4)   // 2048 floats staged per tile (8KB), double buffered
#define MAXK 32
#define TIECAP 32

// ---- order-preserving float <-> uint mapping (ascending) ----
static __device__ __forceinline__ uint32_t fmap_u(float f) {
  uint32_t b = __float_as_uint(f);
  return b ^ ((b & 0x80000000u) ? 0xFFFFFFFFu : 0x80000000u);
}

// ---- JAX threefry-2x32 (20 rounds) ----
static __device__ __forceinline__ void tf_round(uint32_t& x0, uint32_t& x1, int r) {
  x0 += x1; x1 = (x1 << r) | (x1 >> (32 - r)); x1 ^= x0;
}
static __device__ __forceinline__ void threefry2x32(uint32_t k0, uint32_t k1,
                                                    uint32_t c0, uint32_t c1,
                                                    uint32_t& o0, uint32_t& o1) {
  uint32_t k2 = k0 ^ k1 ^ 0x1BD11BDAu;
  uint32_t x0 = c0 + k0, x1 = c1 + k1;
  tf_round(x0,x1,13); tf_round(x0,x1,15); tf_round(x0,x1,26); tf_round(x0,x1,6);
  x0 += k1; x1 += k2 + 1u;
  tf_round(x0,x1,17); tf_round(x0,x1,29); tf_round(x0,x1,16); tf_round(x0,x1,24);
  x0 += k2; x1 += k0 + 2u;
  tf_round(x0,x1,13); tf_round(x0,x1,15); tf_round(x0,x1,26); tf_round(x0,x1,6);
  x0 += k0; x1 += k1 + 3u;
  tf_round(x0,x1,17); tf_round(x0,x1,29); tf_round(x0,x1,16); tf_round(x0,x1,24);
  x0 += k1; x1 += k2 + 4u;
  tf_round(x0,x1,13); tf_round(x0,x1,15); tf_round(x0,x1,26); tf_round(x0,x1,6);
  x0 += k2; x1 += k0 + 5u;
  o0 = x0; o1 = x1;
}

// ---- async global->LDS double-buffered row streamer (CDNA5 path) ----------
// Each lane issues a coalesced b128 async transfer into its private 16B LDS
// slot, waits on its own wave's ASYNCcnt, and consumes exactly the 4 floats
// it deposited. fn(i, rawValue) is called for every element of the row.
template <typename F>
static __device__ __forceinline__ void stream_row(const float* __restrict__ x, int V,
                                                  int tid, float (*stage)[TILE_ELEMS],
                                                  F fn) {
  const int numTiles = V / TILE_ELEMS;
  const int Vmain = numTiles * TILE_ELEMS;
  if (numTiles > 0) {
    uint32_t lds0 = (uint32_t)(uintptr_t)(&stage[0][tid * 4]);
    unsigned long long g0 = (unsigned long long)(uintptr_t)(x + tid * 4);
    asm volatile("global_load_async_to_lds_b128 %0, %1, off" :: "v"(lds0), "v"(g0) : "memory");
  }
  int buf = 0;
  for (int tl = 0; tl < numTiles; ++tl) {
    if (tl + 1 < numTiles) {
      uint32_t lds1 = (uint32_t)(uintptr_t)(&stage[buf ^ 1][tid * 4]);
      unsigned long long g1 =
          (unsigned long long)(uintptr_t)(x + (size_t)(tl + 1) * TILE_ELEMS + tid * 4);
      asm volatile("global_load_async_to_lds_b128 %0, %1, off" :: "v"(lds1), "v"(g1) : "memory");
      asm volatile("s_wait_asynccnt 0x1" ::: "memory");
    } else {
      asm volatile("s_wait_asynccnt 0x0" ::: "memory");
    }
    const float4 vv = *(const float4*)(&stage[buf][tid * 4]);
    const int base = tl * TILE_ELEMS + tid * 4;
    fn(base + 0, vv.x); fn(base + 1, vv.y); fn(base + 2, vv.z); fn(base + 3, vv.w);
    buf ^= 1;
  }
  for (int i = Vmain + tid; i < V; i += NT) fn(i, x[i]);
}

__global__ __launch_bounds__(NT) void sampler_kernel(
    const float* __restrict__ logits,
    const float* __restrict__ temperature,
    const long long* __restrict__ top_k,
    const float* __restrict__ top_p,
    const long long* __restrict__ seeds,
    const long long* __restrict__ pos,
    int K, int B, int V,
    float* __restrict__ out)
{
  const int b   = blockIdx.x;
  const int tid = threadIdx.x;
  const float* __restrict__ x = logits + (size_t)b * (size_t)V;

  __shared__ __align__(16) float stage[2][TILE_ELEMS];
  __shared__ unsigned int histC[256];
  __shared__ float        histS[256];
  __shared__ unsigned int histN[256];
  __shared__ float        redF[NT];
  __shared__ int          redI[NT];
  __shared__ float sM, sZ, sZp, sPtarget, sPacc;
  __shared__ uint32_t sKpref, sPpref, sTu, sU20;
  __shared__ int sKrem, sC, sSampIdx, sRank, sNA, sNT;
  __shared__ uint32_t gA_u[MAXK]; __shared__ int gA_i[MAXK];
  __shared__ int gT_i[TIECAP];

  const float t = temperature[b];
  const bool greedy = (t < 1e-5f);
  const float invT = 1.0f / (greedy ? 1.0f : t);
  int Kc = K; if (Kc < 0) Kc = 0; if (Kc > MAXK) Kc = MAXK;

  // ================= pass 1: fused online max + sum(exp) ===================
  float m = -3.4e38f, s = 0.0f;
  stream_row(x, V, tid, stage, [&](int i, float raw) {
    (void)i;
    float v = raw * invT;
    if (v > m) { s *= expf(m - v); m = v; }
    s += expf(v - m);
  });
  // block combine: M then Z
  redF[tid] = m; __syncthreads();
  for (int off = NT / 2; off > 0; off >>= 1) {
    if (tid < off) redF[tid] = fmaxf(redF[tid], redF[tid + off]);
    __syncthreads();
  }
  const float M = redF[0]; __syncthreads();
  redF[tid] = s * expf(m - M); __syncthreads();
  for (int off = NT / 2; off > 0; off >>= 1) {
    if (tid < off) redF[tid] += redF[tid + off];
    __syncthreads();
  }
  if (tid == 0) { sM = M; sZ = redF[0]; }
  __syncthreads();

  // ============ passes 2-5: combined radix descents (top-k & top-p) ========
  if (tid == 0) {
    long long kk = top_k[b]; if (kk < 1) kk = 1; if (kk > V) kk = V;
    sKrem = (int)kk;
    sPtarget = top_p[b] * sZ;
    sPacc = 0.0f; sKpref = 0u; sPpref = 0u;
  }
  __syncthreads();
  uint32_t kMask = 0u, pMask = 0u;
  for (int level = 0; level < 4; ++level) {
    const int shift = 24 - 8 * level;
    if (tid < 256) { histC[tid] = 0u; histS[tid] = 0.0f; histN[tid] = 0u; }
    __syncthreads();
    const uint32_t kPref = sKpref, pPref = sPpref;
    stream_row(x, V, tid, stage, [&](int i, float raw) {
      (void)i;
      float v = raw * invT; uint32_t u = fmap_u(v);
      unsigned int bin = (u >> shift) & 255u;
      if ((u & kMask) == kPref) atomicAdd(&histC[bin], 1u);
      if ((u & pMask) == pPref) {
        atomicAdd(&histS[bin], expf(v - M));
        atomicAdd(&histN[bin], 1u);
      }
    });
    __syncthreads();
    if (tid == 0) {
      // top-k: bin containing the kRem-th largest
      unsigned int rem = (unsigned int)sKrem, cum = 0u;
      for (int bn = 255; bn >= 0; --bn) {
        unsigned int c = histC[bn];
        if (cum + c >= rem) { sKpref = kPref | ((uint32_t)bn << shift); sKrem = (int)(rem - cum); break; }
        cum += c;
      }
      // top-p: lowest non-empty bin whose exclusive exp-prefix < P
      const float P = sPtarget;
      float acc = sPacc, candAcc = sPacc; int cand = 255;
      for (int bn = 255; bn >= 0; --bn) {
        if (histN[bn] > 0u && acc < P) { cand = bn; candAcc = acc; }
        acc += histS[bn];
      }
      sPpref = pPref | ((uint32_t)cand << shift); sPacc = candAcc;
    }
    __syncthreads();
    kMask |= (0xFFu << shift); pMask |= (0xFFu << shift);
  }
  const uint32_t Tu = (sKpref > sPpref) ? sKpref : sPpref;  // keep = u >= Tu
  if (tid == 0) sTu = Tu;
  __syncthreads();

  // ============ pass 6: gumbel scores + argmax + Z' + kept count ===========
  uint32_t gseed = (uint32_t)((unsigned long long)seeds[b]) ^
                   ((uint32_t)((unsigned long long)pos[b]) * 2654435761u);
  uint32_t fk0, fk1; threefry2x32(0u, 0u, 0u, gseed, fk0, fk1);
  const int H = V >> 1;
  float best = -1.0f; int bidx = 0x7FFFFFFF;
  float zp = 0.0f; int cnt = 0;
  for (int j = tid; j < H; j += NT) {
    uint32_t r0, r1; threefry2x32(fk0, fk1, (uint32_t)j, (uint32_t)(j + H), r0, r1);
    int      idx[2] = { j, j + H };
    uint32_t rr[2]  = { r0, r1 };
    #pragma unroll
    for (int h = 0; h < 2; ++h) {
      int i = idx[h];
      float v = x[i] * invT; uint32_t u = fmap_u(v);
      if (u >= Tu) {
        float e = expf(v - M); zp += e; ++cnt;
        float sc;
        if (greedy) sc = e;
        else {
          float uf = __uint_as_float((rr[h] >> 9) | 0x3F800000u) - 1.0f;
          float q  = -((uf >= 0.9999999403953552f) ? -5.960464477539063e-08f
                                                    : logf(fmaxf(uf, 1e-38f)));
          sc = e / q;
        }
        if (sc > best || (sc == best && i < bidx)) { best = sc; bidx = i; }
      }
    }
  }
  // argmax reduce (first-index tiebreak)
  redF[tid] = best; redI[tid] = bidx; __syncthreads();
  for (int off = NT / 2; off > 0; off >>= 1) {
    if (tid < off) {
      float fb = redF[tid + off]; int ib = redI[tid + off];
      if (fb > redF[tid] || (fb == redF[tid] && ib < redI[tid])) { redF[tid] = fb; redI[tid] = ib; }
    }
    __syncthreads();
  }
  if (tid == 0) sSampIdx = redI[0];
  __syncthreads();
  redF[tid] = zp; redI[tid] = cnt; __syncthreads();
  for (int off = NT / 2; off > 0; off >>= 1) {
    if (tid < off) { redF[tid] += redF[tid + off]; redI[tid] += redI[tid + off]; }
    __syncthreads();
  }
  if (tid == 0) { sZp = redF[0]; sC = redI[0]; }
  __syncthreads();

  // ============ passes 7-10: Kc-th largest among kept (for logprob ids) ====
  const int C = sC;
  if (C > Kc && Kc > 0) {
    if (tid == 0) { sKrem = Kc; sKpref = 0u; }
    __syncthreads();
    uint32_t mask = 0u;
    for (int level = 0; level < 4; ++level) {
      const int shift = 24 - 8 * level;
      if (tid < 256) histC[tid] = 0u;
      __syncthreads();
      const uint32_t pref = sKpref;
      stream_row(x, V, tid, stage, [&](int i, float raw) {
        (void)i;
        float v = raw * invT; uint32_t u = fmap_u(v);
        if (u >= Tu && (u & mask) == pref) atomicAdd(&histC[(u >> shift) & 255u], 1u);
      });
      __syncthreads();
      if (tid == 0) {
        unsigned int rem = (unsigned int)sKrem, cum = 0u;
        for (int bn = 255; bn >= 0; --bn) {
          unsigned int c = histC[bn];
          if (cum + c >= rem) { sKpref = pref | ((uint32_t)bn << shift); sKrem = (int)(rem - cum); break; }
          cum += c;
        }
      }
      __syncthreads();
      mask |= (0xFFu << shift);
    }
    if (tid == 0) sU20 = sKpref;
  } else {
    if (tid == 0) sU20 = Tu;
  }
  __syncthreads();
  const uint32_t u20 = sU20;

  // ============ pass 11: rank count + gather top ids =======================
  const float vs = x[sSampIdx] * invT;
  if (tid == 0) { sNA = 0; sNT = 0; }
  __syncthreads();
  int rk = 0;
  stream_row(x, V, tid, stage, [&](int i, float raw) {
    float v = raw * invT; uint32_t u = fmap_u(v);
    if (v > vs) ++rk;
    if (u >= Tu) {
      if (u > u20) {
        int p = atomicAdd(&sNA, 1);
        if (p < MAXK) { gA_u[p] = u; gA_i[p] = i; }
      } else if (u == u20) {
        int p = atomicAdd(&sNT, 1);
        if (p < TIECAP) gT_i[p] = i;
      }
    }
  });
  redI[tid] = rk; __syncthreads();
  for (int off = NT / 2; off > 0; off >>= 1) {
    if (tid < off) redI[tid] += redI[tid + off];
    __syncthreads();
  }
  if (tid == 0) sRank = redI[0];
  __syncthreads();

  // ============ finalize + write outputs (thread 0) ========================
  if (tid == 0) {
    int nA = sNA; if (nA > MAXK) nA = MAXK;
    int nT_ = sNT; if (nT_ > TIECAP) nT_ = TIECAP;
    // sort gA by (value desc, idx asc)
    for (int a = 1; a < nA; ++a) {
      uint32_t uu = gA_u[a]; int ii = gA_i[a]; int c2 = a - 1;
      while (c2 >= 0 && (gA_u[c2] < uu || (gA_u[c2] == uu && gA_i[c2] > ii))) {
        gA_u[c2 + 1] = gA_u[c2]; gA_i[c2 + 1] = gA_i[c2]; --c2;
      }
      gA_u[c2 + 1] = uu; gA_i[c2 + 1] = ii;
    }
    // sort ties by idx asc (lax.top_k stable tie order)
    for (int a = 1; a < nT_; ++a) {
      int ii = gT_i[a]; int c2 = a - 1;
      while (c2 >= 0 && gT_i[c2] > ii) { gT_i[c2 + 1] = gT_i[c2]; --c2; }
      gT_i[c2 + 1] = ii;
    }
    int ids[1 + MAXK];
    ids[0] = sSampIdx;
    int fill = 0;
    for (int a = 0; a < nA && fill < Kc; ++a) ids[1 + fill++] = gA_i[a];
    for (int a = 0; a < nT_ && fill < Kc; ++a) ids[1 + fill++] = gT_i[a];
    // fill remaining slots with smallest non-kept (-inf) indices
    for (int i = 0; i < V && fill < Kc; ++i) {
      float v = x[i] * invT; uint32_t u = fmap_u(v);
      if (u < sTu) ids[1 + fill++] = i;
    }
    const float logZp = logf(sZp);
    const int stride = Kc + 1;
    out[b] = (float)sSampIdx;                                  // sampled
    for (int j = 0; j < stride; ++j) {
      int id = ids[j];
      out[(size_t)B + (size_t)b * stride + j] = (float)id;     // logprob_token_ids
      float v = x[id] * invT; uint32_t u = fmap_u(v);
      float lp = (u >= sTu) ? (v - sM - logZp) : -INFINITY;
      out[(size_t)B + (size_t)B * stride + (size_t)b * stride + j] = lp;  // logprobs
    }
    out[(size_t)B + 2 * (size_t)B * stride + b] = (float)sRank;           // ranks
  }
}

extern "C" void kernel_launch(void* const* d_in, const int* in_sizes, int n_in,
                              void* d_out, int out_size, void* d_ws, size_t ws_size,
                              hipStream_t stream) {
  (void)n_in; (void)out_size; (void)d_ws; (void)ws_size;
  const float*     logits      = (const float*)d_in[0];
  const float*     temperature = (const float*)d_in[1];
  const long long* top_k       = (const long long*)d_in[2];
  const float*     top_p       = (const float*)d_in[3];
  const long long* seeds       = (const long long*)d_in[4];
  const long long* pos         = (const long long*)d_in[5];
  const int        K           = ((const int*)d_in[6])[0];  // low word works for i32/i64
  const int B = in_sizes[1];
  const int V = in_sizes[0] / B;
  sampler_kernel<<<B, NT, 0, stream>>>(logits, temperature, top_k, top_p,
                                       seeds, pos, K, B, V, (float*)d_out);
}